// GraphSSM_47339129537008
// MI455X (gfx1250) — compile-verified
//
#include <hip/hip_runtime.h>

typedef float v2f __attribute__((ext_vector_type(2)));
typedef float v8f __attribute__((ext_vector_type(8)));

#define TSEQ  64      // sequence length (K of the Toeplitz GEMM)
#define DIMC  128     // channels
#define DST   16      // d_state
#define NT    16      // n-tile per workgroup (WMMA N)
#define DBLK  32      // channels per workgroup
#define PITCH 36      // LDS pitch (floats): 144B keeps 16B align, 16 n-lanes
                      // land on 16 distinct banks (gcd(36,64)=4, period 16)
#define XBUF  (16 * NT * PITCH)   // one 16-timestep chunk in LDS (floats)

#if defined(__gfx1250__) && __has_builtin(__builtin_amdgcn_global_load_async_to_lds_b128)
#define USE_ASYNC 1
// exact param type from hipcc diagnostic: 'int __attribute__((vector_size(16))) *'
typedef int v4i_raw __attribute__((vector_size(16)));
typedef __attribute__((address_space(1))) v4i_raw as1_v4i;   // global side (param 1)
typedef __attribute__((address_space(3))) v4i_raw as3_v4i;   // LDS side    (param 2)
#endif

__device__ __forceinline__ void wait_async0() {
#if __has_builtin(__builtin_amdgcn_s_wait_asynccnt)
    __builtin_amdgcn_s_wait_asynccnt(0);
#else
    asm volatile("s_wait_asynccnt 0x0" ::: "memory");
#endif
}

// ---------------------------------------------------------------------------
// Prepass: build the causal kernel K[d][l] = sum_s C*B*A^l (+D at l=0) and
// expand it into WMMA A-fragment layout:
//   Atab[d][blk = k*4+ti][lane] = { M[16ti+m, 4k+kb2], M[16ti+m, 4k+kb2+1] }
// with m = lane&15, kb2 = (lane>>4)*2, M[t,tin] = K[t-tin] (t>=tin else 0).
// 2 MB total -> L2-resident, reused by all 625 n-tiles.
// ---------------------------------------------------------------------------
__global__ void ssm_build_frags(const float* __restrict__ A, const float* __restrict__ B,
                                const float* __restrict__ C, const float* __restrict__ D,
                                v2f* __restrict__ Atab) {
    const int d   = blockIdx.x;     // 0..127
    const int tid = threadIdx.x;    // 0..255
    __shared__ float Kd[TSEQ];
    if (tid < TSEQ) {
        float acc = 0.f;
#pragma unroll
        for (int s = 0; s < DST; ++s)
            acc += C[d * DST + s] * B[d * DST + s] * __powf(A[d * DST + s], (float)tid);
        if (tid == 0) acc += D[d];
        Kd[tid] = acc;
    }
    __syncthreads();
#pragma unroll
    for (int it = 0; it < 8; ++it) {
        const int idx  = tid + it * 256;       // 0..2047 = 64 blocks * 32 lanes
        const int blk  = idx >> 5;             // k*4 + ti
        const int lane = idx & 31;
        const int k = blk >> 2, ti = blk & 3;
        const int m = lane & 15, kb2 = (lane >> 4) << 1;
        const int lag0 = ti * 16 + m - (k * 4 + kb2);
        v2f a;
        a.x = (lag0 >= 0) ? Kd[lag0]     : 0.f;
        a.y = (lag0 >= 1) ? Kd[lag0 - 1] : 0.f;
        Atab[(size_t)d * 2048 + idx] = a;
    }
}

// ---------------------------------------------------------------------------
// Main: per-channel lower-triangular Toeplitz GEMM via fp32 WMMA.
//   Y_d[64 x N] = M_d[64 x 64] @ X_d[64 x N]
// Grid (N/16, 4). Block 256 = 8 wave32; wave owns 4 channels x 4 t-blocks.
// K streamed in 4 chunks of 16 timesteps, double-buffered in LDS (72 KB).
// All per-lane global addressing uses 32-bit offsets (tensor < 2^31 bytes)
// so loads/stores take the SADDR + 32-bit-voffset form.
// ---------------------------------------------------------------------------
__launch_bounds__(256)
__global__ void ssm_conv_wmma(const float* __restrict__ x,    // [T, N, DIMC]
                              const v2f* __restrict__ Atab,   // [DIMC][64][32]
                              float* __restrict__ y,          // [T, N, DIMC]
                              int N) {
    extern __shared__ float Xs[];   // [2][16 t][NT n][PITCH]

    const int tid   = threadIdx.x;
    const int lane  = tid & 31;
    const int w     = tid >> 5;
    const int n0    = blockIdx.x * NT;
    const int dbase = blockIdx.y * DBLK;

    // staging decode (float4 granularity, 8 lanes cover one 128B channel row)
    const int q  = tid & 7;
    const int nn = (tid >> 3) & 15;
    const int t0 = tid >> 7;

    // WMMA fragment lane decode
    const int fm  = lane & 15;
    const int kb2 = (lane >> 4) << 1;

    // per-channel fragment base pointers: all later offsets are immediates
    const v2f*   abase[4];
    const float* bbase[4];
#pragma unroll
    for (int c = 0; c < 4; ++c) {
        abase[c] = Atab + (size_t)(dbase + w * 4 + c) * 2048 + lane;
        bbase[c] = Xs + kb2 * 576 + fm * PITCH + (w * 4 + c);
    }

    v8f acc[4][4];
#pragma unroll
    for (int c = 0; c < 4; ++c)
#pragma unroll
        for (int ti = 0; ti < 4; ++ti)
            acc[c][ti] = (v8f){0.f, 0.f, 0.f, 0.f, 0.f, 0.f, 0.f, 0.f};

    const unsigned rowstride = (unsigned)(N * DIMC);            // one timestep
    const unsigned xoff0 = (unsigned)((t0 * N + n0 + nn) * DIMC + dbase + q * 4);
    const int      loff0 = (t0 * NT + nn) * PITCH + q * 4;

#ifdef USE_ASYNC
    // ---- async global->LDS staging, double buffered ----
    auto issue_chunk = [&](int kc, int bufsel) {
#pragma unroll
        for (int i = 0; i < 8; ++i) {
            const unsigned goff = xoff0 + (unsigned)(kc * 16 + 2 * i) * rowstride;
            float* l = Xs + loff0 + bufsel * XBUF + i * (2 * NT * PITCH);
            __builtin_amdgcn_global_load_async_to_lds_b128(
                (as1_v4i*)(x + (size_t)goff), (as3_v4i*)l, 0, 0);
        }
    };
    issue_chunk(0, 0);
    wait_async0();
    __syncthreads();
#else
    // ---- register-staged fallback ----
    float4 xr[8];
#pragma unroll
    for (int i = 0; i < 8; ++i)
        xr[i] = *(const float4*)(x + (size_t)(xoff0 + (unsigned)(2 * i) * rowstride));
#endif

#pragma unroll
    for (int kc = 0; kc < 4; ++kc) {
#ifdef USE_ASYNC
        if (kc < 3) issue_chunk(kc + 1, (kc + 1) & 1);   // prefetch next chunk
#else
#pragma unroll
        for (int i = 0; i < 8; ++i)
            *(float4*)(Xs + loff0 + (kc & 1) * XBUF + i * (2 * NT * PITCH)) = xr[i];
        if (kc < 3) {
#pragma unroll
            for (int i = 0; i < 8; ++i)
                xr[i] = *(const float4*)(x + (size_t)(xoff0 +
                            (unsigned)((kc + 1) * 16 + 2 * i) * rowstride));
        }
        __syncthreads();
#endif

        // ---- compute on chunk kc: K-steps k = 4*kc .. 4*kc+3 ----
#pragma unroll
        for (int k4 = 0; k4 < 4; ++k4) {
            const int k = kc * 4 + k4;
#pragma unroll
            for (int c = 0; c < 4; ++c) {
                v2f b;
                b.x = bbase[c][(kc & 1) * XBUF + k4 * 2304];
                b.y = bbase[c][(kc & 1) * XBUF + k4 * 2304 + 576];
#pragma unroll
                for (int ti = 0; ti < 4; ++ti) {
                    if (ti < kc) continue;            // block-triangular skip
                    const v2f a = abase[c][(k * 4 + ti) * 32];
                    acc[c][ti] = __builtin_amdgcn_wmma_f32_16x16x4_f32(
                        false, a, false, b, (short)0, acc[c][ti], false, false);
                }
            }
        }

#ifdef USE_ASYNC
        if (kc < 3) { wait_async0(); __syncthreads(); }
#else
        __syncthreads();
#endif
    }

    // ---- store: pack this wave's 4 consecutive channels into one float4 ----
    const int d0    = dbase + w * 4;
    const int nout  = n0 + fm;
    const int thalf = (lane >> 4) * 8;     // C/D layout: lanes 16-31 hold M=r+8
    const unsigned ybase0 = (unsigned)((thalf * N + nout) * DIMC + d0);
#pragma unroll
    for (int ti = 0; ti < 4; ++ti)
#pragma unroll
        for (int r = 0; r < 8; ++r) {
            const unsigned off = ybase0 + (unsigned)(ti * 16 + r) * rowstride;
            float4 v = make_float4(acc[0][ti][r], acc[1][ti][r],
                                   acc[2][ti][r], acc[3][ti][r]);
            *(float4*)(y + (size_t)off) = v;
        }
}

// ---------------------------------------------------------------------------
extern "C" void kernel_launch(void* const* d_in, const int* in_sizes, int n_in,
                              void* d_out, int out_size, void* d_ws, size_t ws_size,
                              hipStream_t stream) {
    (void)n_in; (void)out_size; (void)ws_size;
    const float* seq = (const float*)d_in[0];   // [T, N, DIMC]
    const float* A   = (const float*)d_in[1];
    const float* B   = (const float*)d_in[2];
    const float* C   = (const float*)d_in[3];
    const float* D   = (const float*)d_in[4];
    float* out = (float*)d_out;

    const int N = in_sizes[0] / (TSEQ * DIMC);  // 10000

    v2f* Atab = (v2f*)d_ws;                     // 128*2048*8B = 2 MB scratch

    ssm_build_frags<<<dim3(DIMC), dim3(256), 0, stream>>>(A, B, C, D, Atab);

    const size_t ldsBytes = (size_t)(2 * XBUF) * sizeof(float);  // 72 KB
    dim3 grid(N / NT, DIMC / DBLK);             // (625, 4)
    ssm_conv_wmma<<<grid, dim3(256), ldsBytes, stream>>>(seq, Atab, out, N);
}